// QFNetBlock_60060822667811
// MI455X (gfx1250) — compile-verified
//
#include <hip/hip_runtime.h>
#include <math.h>

typedef __attribute__((ext_vector_type(2))) float v2f;
typedef __attribute__((ext_vector_type(8))) float v8f;

#define NWIRES 16
#define NSTATE 65536            // 2^16
#define BATCH  32
#define TWO_PI 6.28318530717958647692f

// ---- WMMA f32 16x16x4: D = A(16x4) * B(4x16) + C ----
__device__ __forceinline__ v8f wmma_f32(v2f a, v2f b, v8f c) {
    return __builtin_amdgcn_wmma_f32_16x16x4_f32(
        /*neg_a=*/false, a, /*neg_b=*/false, b,
        /*c_mod=*/(short)0, c, /*reuse_a=*/false, /*reuse_b=*/false);
}

// ---------------------------------------------------------------------------
// Unitary 256-point DFT matrix F[k][j] = exp(+2*pi*i*k*j/256) / 16
// ---------------------------------------------------------------------------
__global__ void init_f_kernel(float* __restrict__ Fre, float* __restrict__ Fim) {
    int k = blockIdx.x;          // 0..255
    int j = threadIdx.x;         // 0..255
    float ang = TWO_PI * (float)((k * j) & 255) * (1.0f / 256.0f);
    float s, c;
    __sincosf(ang, &s, &c);
    Fre[k * 256 + j] = c * 0.0625f;
    Fim[k * 256 + j] = s * 0.0625f;
}

// ---------------------------------------------------------------------------
// Rot(phi,theta,omega) 2x2 entry:
//   [[ep*c, -conj(em)*s],[em*s, conj(ep)*c]]
//   ep = e^{-i(phi+omega)/2}, em = e^{-i(phi-omega)/2}
// ---------------------------------------------------------------------------
__device__ __forceinline__ void rot_entry(const float* __restrict__ rp, int w,
                                          int ra, int rb, float& mr, float& mi) {
    float phi = rp[w * 3 + 0];
    float th  = rp[w * 3 + 1];
    float om  = rp[w * 3 + 2];
    float ch = __cosf(th * 0.5f), sh = __sinf(th * 0.5f);
    float ang;
    if (ra == 0 && rb == 0) { ang = -0.5f * (phi + om); mr =  __cosf(ang) * ch; mi =  __sinf(ang) * ch; }
    else if (ra == 0)       { ang =  0.5f * (phi - om); mr = -__cosf(ang) * sh; mi = -__sinf(ang) * sh; }
    else if (rb == 0)       { ang = -0.5f * (phi - om); mr =  __cosf(ang) * sh; mi =  __sinf(ang) * sh; }
    else                    { ang =  0.5f * (phi + om); mr =  __cosf(ang) * ch; mi =  __sinf(ang) * ch; }
}

// ---------------------------------------------------------------------------
// Build the two 256x256 Kronecker-fused Rot matrices:
//   G01 = kron(R_0..R_7)  (wires 0-7,  state-matrix rows j1)
//   G23 = kron(R_8..R_15) (wires 8-15, state-matrix cols j2)
// ---------------------------------------------------------------------------
__global__ void init_kron_kernel(const float* __restrict__ rp,
                                 float* __restrict__ G01r, float* __restrict__ G01i,
                                 float* __restrict__ G23r, float* __restrict__ G23i) {
    int a = blockIdx.x;      // row 0..255
    int b = threadIdx.x;     // col 0..255
    for (int h = 0; h < 2; ++h) {
        float pr = 1.0f, pi = 0.0f;
        for (int t = 0; t < 8; ++t) {
            int w  = h * 8 + t;
            int ra = (a >> (7 - t)) & 1;
            int rb = (b >> (7 - t)) & 1;
            float mr, mi;
            rot_entry(rp, w, ra, rb, mr, mi);
            float nr = pr * mr - pi * mi;
            float ni = pr * mi + pi * mr;
            pr = nr; pi = ni;
        }
        if (h == 0) { G01r[a * 256 + b] = pr; G01i[a * 256 + b] = pi; }
        else        { G23r[a * 256 + b] = pr; G23i[a * 256 + b] = pi; }
    }
}

// ---------------------------------------------------------------------------
// L2-normalize each batch row and store TRANSPOSED: re[j2*256+j1] = xn[j1*256+j2]
// (so the first DFT pass reads its B operand contiguously). im = 0.
// ---------------------------------------------------------------------------
__global__ void normalize_t_kernel(const float* __restrict__ x,
                                   float* __restrict__ re,
                                   float* __restrict__ im) {
    __shared__ float red[256];
    int b = blockIdx.x;
    const float* xb = x + b * NSTATE;
    float s = 0.0f;
    for (int i = threadIdx.x; i < NSTATE; i += 256) {
        float v = xb[i];
        s += v * v;
    }
    red[threadIdx.x] = s;
    __syncthreads();
    for (int off = 128; off > 0; off >>= 1) {
        if (threadIdx.x < off) red[threadIdx.x] += red[threadIdx.x + off];
        __syncthreads();
    }
    float inv = rsqrtf(red[0]);
    float* rb = re + b * NSTATE;
    float* ib = im + b * NSTATE;
    for (int o = threadIdx.x; o < NSTATE; o += 256) {
        int j2 = o >> 8, j1 = o & 255;
        rb[o] = xb[j1 * 256 + j2] * inv;   // transposed store
        ib[o] = 0.0f;
    }
}

// ---------------------------------------------------------------------------
// Generic complex matmul pass:  D[m][n] = sum_k A[m][k] * S[n][k]   (B = S^T)
// One wave per 16x16 tile; B loads are always contiguous b64.
// TW: multiply D[m][n] by twiddle exp(+2*pi*i*m*n/65536) (fused into DFT-1).
// Used for all four heavy passes (DFT x2, Rot x2) -- each pass transposes,
// so the chain lands back in flat state order.
// ---------------------------------------------------------------------------
template <bool TW>
__global__ void cmatmul_kernel(const float* __restrict__ Ar, const float* __restrict__ Ai,
                               const float* __restrict__ Sr, const float* __restrict__ Si,
                               float* __restrict__ Dr, float* __restrict__ Di) {
    int lane = threadIdx.x;                 // 0..31, one wave, EXEC all-ones
    int tile = blockIdx.x & 255;            // 16x16 grid of 16x16 tiles
    int b    = blockIdx.x >> 8;             // batch
    int tm = tile >> 4, tn = tile & 15;
    int am = tm * 16 + (lane & 15);         // A: M = lane%16
    int kg = (lane >> 4) * 2;               // lanes 16-31 hold K+2,K+3
    int nI = tn * 16 + (lane & 15);         // B/D: N = lane%16
    const float* ar = Ar + am * 256;
    const float* ai = Ai + am * 256;
    const float* sr = Sr + b * NSTATE + nI * 256;
    const float* si = Si + b * NSTATE + nI * 256;

    v8f rr = {}, ii = {}, ri = {}, ir = {};
#pragma unroll 4
    for (int k = 0; k < 256; k += 4) {
        int ka = k + kg;
        v2f aR = *(const v2f*)(ar + ka);
        v2f aI = *(const v2f*)(ai + ka);
        v2f bR = *(const v2f*)(sr + ka);
        v2f bI = *(const v2f*)(si + ka);
        rr = wmma_f32(aR, bR, rr);
        ii = wmma_f32(aI, bI, ii);
        ri = wmma_f32(aR, bI, ri);
        ir = wmma_f32(aI, bR, ir);
    }

    float* dr = Dr + b * NSTATE;
    float* di = Di + b * NSTATE;
    int m0 = tm * 16 + (lane >> 4) * 8;     // C/D: VGPR v -> M = v + 8*(lane/16)
#pragma unroll
    for (int v = 0; v < 8; ++v) {
        int m = m0 + v;
        float re  = rr[v] - ii[v];
        float im2 = ri[v] + ir[v];
        if constexpr (TW) {
            float ang = TWO_PI * (float)((m * nI) & 65535) * (1.0f / 65536.0f);
            float s, c;
            __sincosf(ang, &s, &c);
            float tr = re * c - im2 * s;
            float ti = re * s + im2 * c;
            re = tr; im2 = ti;
        }
        dr[m * 256 + nI] = re;
        di[m * 256 + nI] = im2;
    }
}

// ---------------------------------------------------------------------------
// CNOT ring composed into one permutation, fused with |.| output.
// Gate i (i=0..15, applied first-to-last): control wire i, target (i+1)%16;
// wire w lives at flat bit (15-w).  final[j] = s0[pi_0(pi_1(...pi_15(j)))].
// ---------------------------------------------------------------------------
__global__ void cnot_abs_kernel(const float* __restrict__ Sre,
                                const float* __restrict__ Sim,
                                float* __restrict__ out) {
    int tid = blockIdx.x * 256 + threadIdx.x;
    int b = tid >> 16;
    int j = tid & (NSTATE - 1);
    int idx = j;
#pragma unroll
    for (int i = NWIRES - 1; i >= 0; --i) {
        int cb = 15 - i;                 // control bit
        int tb = 15 - ((i + 1) & 15);    // target bit
        if ((idx >> cb) & 1) idx ^= (1 << tb);
    }
    float re = Sre[b * NSTATE + idx];
    float im = Sim[b * NSTATE + idx];
    out[tid] = sqrtf(re * re + im * im);
}

// ---------------------------------------------------------------------------
extern "C" void kernel_launch(void* const* d_in, const int* in_sizes, int n_in,
                              void* d_out, int out_size, void* d_ws, size_t ws_size,
                              hipStream_t stream) {
    (void)in_sizes; (void)n_in; (void)out_size; (void)ws_size;
    const float* x  = (const float*)d_in[0];   // (32, 512, 128) f32
    const float* rp = (const float*)d_in[1];   // (1, 16, 3)     f32

    float* ws = (float*)d_ws;
    const size_t NS = (size_t)BATCH * NSTATE;  // 2,097,152 floats (8 MB each)
    float* Are  = ws;
    float* Aim  = Are  + NS;
    float* Bre  = Aim  + NS;
    float* Bim  = Bre  + NS;
    float* Fre  = Bim  + NS;
    float* Fim  = Fre  + 65536;
    float* G01r = Fim  + 65536;
    float* G01i = G01r + 65536;
    float* G23r = G01i + 65536;
    float* G23i = G23r + 65536;

    init_f_kernel     <<<256, 256, 0, stream>>>(Fre, Fim);
    init_kron_kernel  <<<256, 256, 0, stream>>>(rp, G01r, G01i, G23r, G23i);
    normalize_t_kernel<<<BATCH, 256, 0, stream>>>(x, Are, Aim);   // A = X^T

    // QFT == unitary 65536-pt DFT, four-step (256 x 256):
    //   B = twiddle .* (F * A^T)   ;   A = F * B^T   (now flat state order)
    cmatmul_kernel<true> <<<8192, 32, 0, stream>>>(Fre, Fim, Are, Aim, Bre, Bim);
    cmatmul_kernel<false><<<8192, 32, 0, stream>>>(Fre, Fim, Bre, Bim, Are, Aim);

    // VQC Rot layer: kron(R0..R15) = kron(G01, G23);  S' = G01 * S * G23^T
    //   B = G23 * A^T   ;   A = G01 * B^T   (back in flat state order)
    cmatmul_kernel<false><<<8192, 32, 0, stream>>>(G23r, G23i, Are, Aim, Bre, Bim);
    cmatmul_kernel<false><<<8192, 32, 0, stream>>>(G01r, G01i, Bre, Bim, Are, Aim);

    // CNOT ring permutation + abs -> output:
    cnot_abs_kernel<<<8192, 256, 0, stream>>>(Are, Aim, (float*)d_out);
}